// QAOA_40939628265592
// MI455X (gfx1250) — compile-verified
//
#include <hip/hip_runtime.h>

// QAOA expectation value, MI455X (gfx1250, wave32).
// Work is tiny (H=6, S=14, N=4): one wave does everything in registers.
// The final weighted reduction Sum_{t,s} w_t*temp[t,s] is performed with
// V_WMMA_F32_16X16X4_F32 (K-chained x4 over the s dimension), using an
// all-ones B operand and a sum-of-all-D-entries/16 readout so the result is
// invariant to the exact A/B/D VGPR layouts.
//
// Gray states: the reference's "flip bit ctz(i) at step i" walk is exactly the
// binary-reflected Gray code, so gray[s] == s ^ (s >> 1) — computed inline
// instead of a variably-indexed private array (avoids a cndmask select chain).

typedef __attribute__((ext_vector_type(2))) float v2f;
typedef __attribute__((ext_vector_type(8))) float v8f;

#define H_TERMS 6   // Hamiltonian terms
#define NQ      4   // qubits
#define S_ST    14  // Gray-code states: 2*D*(H+1)

struct Cplx { float re, im; };
__device__ __forceinline__ Cplx cmul(Cplx a, Cplx b) {
    return Cplx{ a.re * b.re - a.im * b.im, a.re * b.im + a.im * b.re };
}

__global__ __launch_bounds__(32)
void qaoa_wmma_kernel(const float* __restrict__ weights,
                      const int*   __restrict__ flags,   // [H][N], 0/1
                      const float* __restrict__ beta,
                      const float* __restrict__ gamma,
                      float*       __restrict__ out) {   // out[0]=re, out[1]=im
    const int lane = threadIdx.x;              // 0..31, single wave32

    const float b0 = beta[0];
    const float g0 = gamma[0];

    float w[H_TERMS];
    int   fl[H_TERMS][NQ];
    for (int j = 0; j < H_TERMS; ++j) {
        w[j] = weights[j];
        for (int n = 0; n < NQ; ++n) fl[j][n] = (flags[j * NQ + n] != 0);
    }

    float cth[H_TERMS], sth[H_TERMS];
    for (int j = 0; j < H_TERMS; ++j) {
        float th = g0 * w[j];
        cth[j] = cosf(th);
        sth[j] = sinf(th);
    }
    const float cb = cosf(b0);

    // A-element mapping (any bijection onto 16x16 (t,s) works for the final
    // scalar): t = lane&15 ; s = 4*block + (lane>>4)*2 + comp.
    const int t   = lane & 15;
    const int khi = (lane >> 4) * 2;

    v8f acc_re = {0.f, 0.f, 0.f, 0.f, 0.f, 0.f, 0.f, 0.f};
    v8f acc_im = {0.f, 0.f, 0.f, 0.f, 0.f, 0.f, 0.f, 0.f};
    const v2f bones = {1.0f, 1.0f};            // B = all-ones (layout invariant)

    for (int blk = 0; blk < 4; ++blk) {
        v2f a_re, a_im;
        for (int comp = 0; comp < 2; ++comp) {
            const int s = 4 * blk + khi + comp;
            float vre = 0.f, vim = 0.f;
            if (t < H_TERMS && s < S_ST) {
                // Binary-reflected Gray code == reference's ctz-flip walk.
                const unsigned gs = (unsigned)s ^ ((unsigned)s >> 1);

                // s_h factors: prod_j (hfl ? -i*sin(theta_j) : cos(theta_j))
                Cplx sh1 = {1.f, 0.f}, sh2 = {1.f, 0.f};
                for (int j = 0; j < H_TERMS; ++j) {
                    Cplx c1 = ((gs >> (1 + j)) & 1u) ? Cplx{0.f, -sth[j]} : Cplx{cth[j], 0.f};
                    Cplx c2 = ((gs >> (8 + j)) & 1u) ? Cplx{0.f, -sth[j]} : Cplx{cth[j], 0.f};
                    sh1 = cmul(sh1, c1);
                    sh2 = cmul(sh2, c2);
                }

                // Fb diagonal: PauliX branch has zero diagonal -> kills trace.
                const int   xf1 = (gs >> 0) & 1u;
                const int   xf2 = (gs >> 7) & 1u;
                const float f1d = xf1 ? 0.f : cb;
                const float f2d = xf2 ? 0.f : cb;

                const Cplx s12 = cmul(sh1, sh2);

                // temp[t,s] = prod_n trace(base . F1 . UNIF . F2)  (elementwise,
                // only diagonals survive since base off-diag == 0).
                Cplx temp = {1.f, 0.f};
                for (int n = 0; n < NQ; ++n) {
                    int c1n = 0, c2n = 0;
                    for (int j = 0; j < H_TERMS; ++j) {
                        c1n += (int)((gs >> (1 + j)) & 1u) & fl[j][n];
                        c2n += (int)((gs >> (8 + j)) & 1u) & fl[j][n];
                    }
                    const float z1 = (c1n & 1) ? -1.f : 1.f;   // Zp1[1][1]
                    const float z2 = (c2n & 1) ? -1.f : 1.f;   // Zp2[1][1]
                    const float bz = fl[t][n] ? -1.f : 1.f;    // base[1][1]
                    // trace = s12 * f1d*0.5*f2d * (1 + bz*z1*z2)
                    const float scale = f1d * 0.5f * f2d * (1.f + bz * z1 * z2);
                    const Cplx tr = { s12.re * scale, s12.im * scale };
                    temp = cmul(temp, tr);
                }
                vre = w[t] * temp.re;
                vim = w[t] * temp.im;
            }
            a_re[comp] = vre;
            a_im[comp] = vim;
        }
        // D = A x Ones + C ; chained over 4 K-blocks covering s = 0..15.
        acc_re = __builtin_amdgcn_wmma_f32_16x16x4_f32(
            false, a_re, false, bones, (short)0, acc_re, false, false);
        acc_im = __builtin_amdgcn_wmma_f32_16x16x4_f32(
            false, a_im, false, bones, (short)0, acc_im, false, false);
    }

    // Sum ALL 256 entries of D (8 VGPRs x 32 lanes) == 16 * (sum of all A
    // entries) == 16 * answer, regardless of the hardware's D layout.
    float pre = 0.f, pim = 0.f;
    for (int r = 0; r < 8; ++r) { pre += acc_re[r]; pim += acc_im[r]; }
    for (int off = 16; off > 0; off >>= 1) {
        pre += __shfl_xor(pre, off, 32);
        pim += __shfl_xor(pim, off, 32);
    }
    if (lane == 0) {
        out[0] = pre * 0.0625f;   // /16, exact in f32
        out[1] = pim * 0.0625f;
    }
}

extern "C" void kernel_launch(void* const* d_in, const int* in_sizes, int n_in,
                              void* d_out, int out_size, void* d_ws, size_t ws_size,
                              hipStream_t stream) {
    (void)in_sizes; (void)n_in; (void)out_size; (void)d_ws; (void)ws_size;
    const float* weights = (const float*)d_in[0];   // [6]  f32
    const int*   flags   = (const int*)  d_in[1];   // [6,4] bool->int
    const float* beta    = (const float*)d_in[2];   // [1]  f32
    const float* gamma   = (const float*)d_in[3];   // [1]  f32
    float* out = (float*)d_out;                      // complex64 scalar -> 2 floats

    qaoa_wmma_kernel<<<1, 32, 0, stream>>>(weights, flags, beta, gamma, out);
}